// DynamicEmbedding_66494683677006
// MI455X (gfx1250) — compile-verified
//
#include <hip/hip_runtime.h>

typedef __attribute__((ext_vector_type(2))) float v2f;
typedef __attribute__((ext_vector_type(8))) float v8f;

#define VOCAB      32000
#define D_EMB      300
#define D_MODEL    512
#define ROWS_TOTAL 1024            // 8 * 128 tokens
#define TILE_M     16
#define WAVES_PB   8
#define TILE_N     (WAVES_PB * 16) // 128
#define LDS_STRIDE 308             // conflict-free across 16 lanes, 8B-aligned K entry

__global__ __launch_bounds__(256) void emb_fc_wmma_f32(
    const int*   __restrict__ tokens,   // [1024]
    const float* __restrict__ weights,  // [VOCAB, 300]
    const float* __restrict__ fcw,      // [512, 300]
    const float* __restrict__ fcb,      // [512]
    float*       __restrict__ out)      // [1024, 512]
{
    __shared__ int   s_tok[TILE_M];
    __shared__ float s_emb[TILE_M * LDS_STRIDE];

    const int tid  = threadIdx.x;
    const int wave = tid >> 5;
    const int lane = tid & 31;
    const int row_base = blockIdx.y * TILE_M;
    const int col_base = blockIdx.x * TILE_N + wave * 16;

    // --- gather token ids for this block's 16 rows ---
    if (tid < TILE_M) s_tok[tid] = tokens[row_base + tid];
    __syncthreads();

    // --- cooperative gather of 16 embedding rows into LDS (float2 granularity) ---
    // 16 rows * 150 float2 = 2400 elements, 256 threads
    for (int idx = tid; idx < TILE_M * (D_EMB / 2); idx += 256) {
        int r  = idx / (D_EMB / 2);
        int c2 = idx - r * (D_EMB / 2);
        const float* src = weights + (size_t)s_tok[r] * D_EMB + (size_t)c2 * 2;
        v2f v = *(const v2f*)src;
        *(v2f*)&s_emb[r * LDS_STRIDE + c2 * 2] = v;
    }
    __syncthreads();

    // --- WMMA fragment addressing (ISA 16x4 f32 A / 4x16 B layout) ---
    const int half = lane >> 4;      // 0 -> K = {k, k+1}; 1 -> K = {k+2, k+3}
    const int l15  = lane & 15;      // A: row index, B: column index
    const float* brow = fcw + (size_t)(col_base + l15) * D_EMB;

    v8f acc = {};
    #pragma unroll 5
    for (int k = 0; k < D_EMB; k += 4) {   // 300 / 4 = 75 exact steps
        int kk = k + half * 2;
        v2f a = *(const v2f*)&s_emb[l15 * LDS_STRIDE + kk];  // ds_load_b64
        v2f b = *(const v2f*)(brow + kk);                    // global_load_b64
        // (neg_a, A, neg_b, B, c_mod, C, reuse_a, reuse_b)
        acc = __builtin_amdgcn_wmma_f32_16x16x4_f32(
                  false, a, false, b, (short)0, acc, false, false);
    }

    // --- bias (varies along N only) + store D tile ---
    const float bias = fcb[col_base + l15];
    const int   m0   = row_base + half * 8;
    float* outp = out + (size_t)m0 * D_MODEL + col_base + l15;
    #pragma unroll
    for (int i = 0; i < 8; ++i) {
        outp[(size_t)i * D_MODEL] = acc[i] + bias;
    }
}

extern "C" void kernel_launch(void* const* d_in, const int* in_sizes, int n_in,
                              void* d_out, int out_size, void* d_ws, size_t ws_size,
                              hipStream_t stream)
{
    const int*   tokens  = (const int*)  d_in[0];  // (8,128) int32
    const float* weights = (const float*)d_in[1];  // (32000,300) f32
    const float* fcw     = (const float*)d_in[2];  // (512,300) f32
    const float* fcb     = (const float*)d_in[3];  // (512,) f32
    float*       out     = (float*)d_out;          // (8,128,512) f32

    dim3 grid(D_MODEL / TILE_N, ROWS_TOTAL / TILE_M); // (4, 64)
    emb_fc_wmma_f32<<<grid, 256, 0, stream>>>(tokens, weights, fcw, fcb, out);
}